// MultiHeadGATLayer_36558761623740
// MI455X (gfx1250) — compile-verified
//
#include <hip/hip_runtime.h>
#include <hip/hip_bf16.h>

#define NN   50000
#define EE   800000
#define DIN  256
#define DOUT 64
#define HH   4
#define NEG_SLOPE 0.01f

typedef __attribute__((ext_vector_type(2))) float v2f;
typedef __attribute__((ext_vector_type(8))) float v8f;

// ---------------------------------------------------------------------------
// Hardware f32 atomic add (GLOBAL_ATOMIC_ADD_F32, no return -> STOREcnt).
// Inline asm guarantees the CDNA5 instruction instead of a CAS loop.
// ---------------------------------------------------------------------------
__device__ __forceinline__ void atomAddF32(float* p, float v) {
  asm volatile("global_atomic_add_f32 %0, %1, off" :: "v"(p), "v"(v) : "memory");
}

// Order-preserving float<->uint mapping so segment-max can use u32 atomics.
__device__ __forceinline__ unsigned f2key(float f) {
  unsigned b = __float_as_uint(f);
  return (b & 0x80000000u) ? ~b : (b | 0x80000000u);
}
__device__ __forceinline__ float key2f(unsigned k) {
  unsigned b = (k & 0x80000000u) ? (k ^ 0x80000000u) : ~k;
  return __uint_as_float(b);
}

// ---------------------------------------------------------------------------
// Zero-fill kernel (graph-capture safe; d_out/d_ws are poisoned by harness).
// ---------------------------------------------------------------------------
__global__ void zero_kernel(float* __restrict__ p, int n) {
  int i = blockIdx.x * blockDim.x + threadIdx.x;
  if (i < n) p[i] = 0.0f;
}

// ---------------------------------------------------------------------------
// Kernel 1: z[hd][n][o] = sum_k h[n][k] * W[hd][o][k]    (V_WMMA_F32_16X16X4_F32)
// Block = 128 threads = 4 waves; wave w handles head w; each wave computes a
// 16-row x 64-col tile (4 accumulators of 16x16), K-loop over 256 in steps of 4.
// A frag (16x4 f32): lane L: row M = L&15; VGPR0/1 hold K = 2*(L>>4) + {0,1}.
// B frag (4x16 f32): lane L: col N = L&15; VGPR0/1 hold K = 2*(L>>4) + {0,1}.
// D (16x16 f32): lane L: col N = L&15; VGPR r holds row M = r + 8*(L>>4).
// ---------------------------------------------------------------------------
__global__ __launch_bounds__(128)
void gat_gemm_wmma(const float* __restrict__ hmat, const float* __restrict__ W,
                   float* __restrict__ z) {
  const int hd   = threadIdx.x >> 5;   // wave index == head
  const int lane = threadIdx.x & 31;
  const int hi   = lane >> 4;          // 0 or 1
  const int lo   = lane & 15;
  const int row_base = blockIdx.x * 16;

  const float* __restrict__ Wh   = W + (size_t)hd * DOUT * DIN;
  const float* __restrict__ Arow = hmat + (size_t)(row_base + lo) * DIN + 2 * hi;
  const float* __restrict__ Bcol = Wh + (size_t)lo * DIN + 2 * hi;

  v8f acc0 = {}, acc1 = {}, acc2 = {}, acc3 = {};

  #pragma unroll 4
  for (int kb = 0; kb < DIN; kb += 4) {
    v2f a  = *(const v2f*)(Arow + kb);
    v2f b0 = *(const v2f*)(Bcol + kb);
    v2f b1 = *(const v2f*)(Bcol + kb + 16 * DIN);
    v2f b2 = *(const v2f*)(Bcol + kb + 32 * DIN);
    v2f b3 = *(const v2f*)(Bcol + kb + 48 * DIN);
    acc0 = __builtin_amdgcn_wmma_f32_16x16x4_f32(false, a, false, b0, (short)0, acc0, false, false);
    acc1 = __builtin_amdgcn_wmma_f32_16x16x4_f32(false, a, false, b1, (short)0, acc1, false, false);
    acc2 = __builtin_amdgcn_wmma_f32_16x16x4_f32(false, a, false, b2, (short)0, acc2, false, false);
    acc3 = __builtin_amdgcn_wmma_f32_16x16x4_f32(false, a, false, b3, (short)0, acc3, false, false);
  }

  float* __restrict__ zp = z + ((size_t)hd * NN + row_base) * DOUT;
  #pragma unroll
  for (int r = 0; r < 8; ++r) {
    const size_t mrow = (size_t)(r + 8 * hi) * DOUT;
    zp[mrow +  0 + lo] = acc0[r];
    zp[mrow + 16 + lo] = acc1[r];
    zp[mrow + 32 + lo] = acc2[r];
    zp[mrow + 48 + lo] = acc3[r];
  }
}

// ---------------------------------------------------------------------------
// Kernel 2: e_src[hd][n] = dot(z[hd][n], att[hd][0:64]);
//           e_dst[hd][n] = dot(z[hd][n], att[hd][64:128])
// One thread per (hd, n); idx == hd*NN + n matches z's row layout.
// ---------------------------------------------------------------------------
__global__ void gat_escore(const float* __restrict__ z, const float* __restrict__ att,
                           float* __restrict__ es, float* __restrict__ ed) {
  int idx = blockIdx.x * blockDim.x + threadIdx.x;
  if (idx >= HH * NN) return;
  const int hd = idx / NN;
  const float* __restrict__ zr = z + (size_t)idx * DOUT;
  const float* __restrict__ as = att + hd * 2 * DOUT;
  const float* __restrict__ ad = as + DOUT;
  float s0 = 0.0f, s1 = 0.0f;
  #pragma unroll 8
  for (int o = 0; o < DOUT; ++o) {
    float v = zr[o];
    s0 += v * as[o];
    s1 += v * ad[o];
  }
  es[idx] = s0;
  ed[idx] = s1;
}

// ---------------------------------------------------------------------------
// Kernel 3: segment max over dst of leakyrelu(e_src[src] + e_dst[dst]),
// via u32 atomicMax on order-preserving keys (init key = 0 == "-NaN" == bottom).
// ---------------------------------------------------------------------------
__global__ void gat_edge_max(const int* __restrict__ src, const int* __restrict__ dst,
                             const float* __restrict__ es, const float* __restrict__ ed,
                             unsigned* __restrict__ mkey) {
  int e = blockIdx.x * blockDim.x + threadIdx.x;
  if (e >= EE) return;
  const int s = src[e], d = dst[e];
  #pragma unroll
  for (int hd = 0; hd < HH; ++hd) {
    float v = es[hd * NN + s] + ed[hd * NN + d];
    v = (v > 0.0f) ? v : NEG_SLOPE * v;
    atomicMax(&mkey[hd * NN + d], f2key(v));
  }
}

// Kernel 4: decode keys -> float max, non-finite (untouched nodes) -> 0.
__global__ void gat_max_fin(unsigned* __restrict__ mkey) {
  int i = blockIdx.x * blockDim.x + threadIdx.x;
  if (i >= HH * NN) return;
  float mv = key2f(mkey[i]);
  if (!isfinite(mv)) mv = 0.0f;
  ((float*)mkey)[i] = mv;
}

// ---------------------------------------------------------------------------
// Kernel 5: ex = exp(e - m[dst]); den[dst] += ex  (hardware f32 atomic add).
// One thread per (hd, e); layout t = hd*EE + e -> coalesced src/dst/ex access.
// ---------------------------------------------------------------------------
__global__ void gat_edge_exp(const int* __restrict__ src, const int* __restrict__ dst,
                             const float* __restrict__ es, const float* __restrict__ ed,
                             const float* __restrict__ m, float* __restrict__ ex,
                             float* __restrict__ den) {
  int t = blockIdx.x * blockDim.x + threadIdx.x;
  if (t >= HH * EE) return;
  const int hd = t / EE;
  const int e  = t - hd * EE;
  const int s = src[e], d = dst[e];
  float v = es[hd * NN + s] + ed[hd * NN + d];
  v = (v > 0.0f) ? v : NEG_SLOPE * v;
  float x = __expf(v - m[hd * NN + d]);
  ex[t] = x;
  atomAddF32(&den[hd * NN + d], x);
}

// ---------------------------------------------------------------------------
// Kernel 6: out[dst][hd*64+o] += (ex/den[dst]) * z[hd][src][o]
// One wave per (edge, head): 32 lanes x float2 = 64 outputs, fully coalesced
// 256B gather from L2-resident z, then 64 GLOBAL_ATOMIC_ADD_F32 scatters.
// ---------------------------------------------------------------------------
__global__ void gat_scatter(const int* __restrict__ src, const int* __restrict__ dst,
                            const float* __restrict__ z, const float* __restrict__ ex,
                            const float* __restrict__ den, float* __restrict__ out) {
  int t = blockIdx.x * blockDim.x + threadIdx.x;   // E*H*32 = 102.4M < 2^31
  const int o2 = t & 31;
  const int eh = t >> 5;
  if (eh >= EE * HH) return;
  const int hd = eh & 3;
  const int e  = eh >> 2;
  const int s = src[e], d = dst[e];
  const float a = ex[hd * EE + e] / fmaxf(den[hd * NN + d], 1e-16f);
  const float2 v = *(const float2*)(z + ((size_t)hd * NN + s) * DOUT + o2 * 2);
  float* op = out + (size_t)d * (HH * DOUT) + hd * DOUT + o2 * 2;
  atomAddF32(op,     v.x * a);
  atomAddF32(op + 1, v.y * a);
}

// ---------------------------------------------------------------------------
extern "C" void kernel_launch(void* const* d_in, const int* in_sizes, int n_in,
                              void* d_out, int out_size, void* d_ws, size_t ws_size,
                              hipStream_t stream) {
  (void)in_sizes; (void)n_in; (void)out_size; (void)ws_size;

  const float* hmat = (const float*)d_in[0];   // (N, 256)
  const int*   src  = (const int*)  d_in[1];   // (E,)
  const int*   dst  = (const int*)  d_in[2];   // (E,)
  const float* W    = (const float*)d_in[3];   // (H, 64, 256)
  const float* att  = (const float*)d_in[4];   // (H, 128)
  float*       out  = (float*)d_out;           // (N, H*64)

  // Workspace layout (floats/uints, 4B each):
  float*    z    = (float*)d_ws;                         // H*N*64 = 12,800,000
  float*    es   = z  + (size_t)HH * NN * DOUT;          // H*N    =    200,000
  float*    ed   = es + (size_t)HH * NN;                 // H*N
  unsigned* mkey = (unsigned*)(ed + (size_t)HH * NN);    // H*N (keys, then floats)
  float*    den  = (float*)(mkey + (size_t)HH * NN);     // H*N
  float*    ex   = den + (size_t)HH * NN;                // H*E  = 3,200,000

  const int HN  = HH * NN;          //   200,000
  const int OUT = NN * HH * DOUT;   // 12,800,000

  // Init: out = 0, max-keys = 0 (== bottom), den = 0.
  zero_kernel<<<(OUT + 255) / 256, 256, 0, stream>>>(out, OUT);
  zero_kernel<<<(HN  + 255) / 256, 256, 0, stream>>>((float*)mkey, HN);
  zero_kernel<<<(HN  + 255) / 256, 256, 0, stream>>>(den, HN);

  // 1) z = h @ W^T per head (WMMA f32).
  gat_gemm_wmma<<<NN / 16, 128, 0, stream>>>(hmat, W, z);

  // 2) attention logit halves.
  gat_escore<<<(HN + 255) / 256, 256, 0, stream>>>(z, att, es, ed);

  // 3) segment max over dst.
  gat_edge_max<<<(EE + 255) / 256, 256, 0, stream>>>(src, dst, es, ed, mkey);
  gat_max_fin<<<(HN + 255) / 256, 256, 0, stream>>>(mkey);

  // 4) exp + denominator.
  gat_edge_exp<<<(HH * EE + 255) / 256, 256, 0, stream>>>(src, dst, es, ed,
                                                          (const float*)mkey, ex, den);

  // 5) weighted scatter-add into output.
  const int SCAT = EE * HH * 32;    // 102,400,000 threads
  gat_scatter<<<(SCAT + 255) / 256, 256, 0, stream>>>(src, dst, z, ex, den, out);
}